// Patcher_68496138436885
// MI455X (gfx1250) — compile-verified
//
#include <hip/hip_runtime.h>
#include <cstdint>

typedef float v4f __attribute__((ext_vector_type(4)));
typedef float v2f __attribute__((ext_vector_type(2)));

#define THREADS 256

// 3-stage 2x2x2 Haar butterfly on one (2D x 2H x 4W) input brick -> 8 subbands x 2 w.
// Net scale (1/sqrt2)^3 * 2*sqrt2 == 1. Subband s = 4*sD + 2*sH + sW, channel = s*16+c,
// so per-subband output stride = 16*131072 = 2097152 elements.
__device__ __forceinline__ void haar_butterfly_store(const v4f a00, const v4f a01,
                                                     const v4f a10, const v4f a11,
                                                     float* __restrict__ o)
{
    // Stage 1: W butterfly within each quad
    const v2f wl00 = { a00.x + a00.y, a00.z + a00.w };
    const v2f wh00 = { a00.x - a00.y, a00.z - a00.w };
    const v2f wl01 = { a01.x + a01.y, a01.z + a01.w };
    const v2f wh01 = { a01.x - a01.y, a01.z - a01.w };
    const v2f wl10 = { a10.x + a10.y, a10.z + a10.w };
    const v2f wh10 = { a10.x - a10.y, a10.z - a10.w };
    const v2f wl11 = { a11.x + a11.y, a11.z + a11.w };
    const v2f wh11 = { a11.x - a11.y, a11.z - a11.w };

    // Stage 2: H butterfly (j=0 vs j=1), for d-even (0) and d-odd (1)
    const v2f pLL0 = wl00 + wl01, pHL0 = wl00 - wl01;
    const v2f pLH0 = wh00 + wh01, pHH0 = wh00 - wh01;
    const v2f pLL1 = wl10 + wl11, pHL1 = wl10 - wl11;
    const v2f pLH1 = wh10 + wh11, pHH1 = wh10 - wh11;

    // Stage 3: D butterfly -> 8 subbands, streamed non-temporally (1 GB stream >> 192 MB L2)
    __builtin_nontemporal_store(pLL0 + pLL1, (v2f*)(o + 0u * 2097152u)); // LLL
    __builtin_nontemporal_store(pLH0 + pLH1, (v2f*)(o + 1u * 2097152u)); // LLH
    __builtin_nontemporal_store(pHL0 + pHL1, (v2f*)(o + 2u * 2097152u)); // LHL
    __builtin_nontemporal_store(pHH0 + pHH1, (v2f*)(o + 3u * 2097152u)); // LHH
    __builtin_nontemporal_store(pLL0 - pLL1, (v2f*)(o + 4u * 2097152u)); // HLL
    __builtin_nontemporal_store(pLH0 - pLH1, (v2f*)(o + 5u * 2097152u)); // HLH
    __builtin_nontemporal_store(pHL0 - pHL1, (v2f*)(o + 6u * 2097152u)); // HHL
    __builtin_nontemporal_store(pHH0 - pHH1, (v2f*)(o + 7u * 2097152u)); // HHH
}

// x:   [8, 16, 64, 128, 128] f32   (B, C, D, H, W)
// out: [8, 128, 32, 64, 64]  f32   (B, 8C, D/2, H/2, W/2), subband-major channels
// Each thread handles one (b, c, d, wq) at TWO h-slices: h and h+32.
__global__ __launch_bounds__(THREADS, 1)
void haar3_fwd_kernel(const float* __restrict__ x, float* __restrict__ out)
{
    __shared__ v4f tile[THREADS * 8];   // 32 KB: 8 x float4 per thread

    const unsigned tid = threadIdx.x;
    const unsigned gid = blockIdx.x * THREADS + tid;

    const unsigned wq = gid & 31u;            // lane-contiguous => 512 B/wave per load
    unsigned t = gid >> 5;
    const unsigned h = t & 31u; t >>= 5;      // output h; also handles h+32
    const unsigned d = t & 31u; t >>= 5;      // output d (input 2d, 2d+1)
    const unsigned c = t & 15u;
    const unsigned b = t >> 4;

    // input strides (elements): W=1, H=128, D=16384, C=1048576, B=16777216
    const unsigned in_off = ((b * 16u + c) * 64u + 2u * d) * 16384u
                          + (2u * h) * 128u + 4u * wq;
    const float* gA = x + in_off;             // h-slice A: (2h)
    const float* gB = gA + 8192u;             // h-slice B: (2h+64) == 2*(h+32)

    // per-thread LDS slot; flat-addr low 32 bits == LDS byte address (ISA 10.2)
    v4f* slot = &tile[tid * 8u];
    const unsigned l0 = (unsigned)(uintptr_t)slot;

    // CDNA5 async copies (ASYNCcnt): 8 in flight before the single wait.
    asm volatile("global_load_async_to_lds_b128 %0, %1, off" :: "v"(l0),        "v"(gA)          : "memory");
    asm volatile("global_load_async_to_lds_b128 %0, %1, off" :: "v"(l0 +  16u), "v"(gA + 128u)   : "memory");
    asm volatile("global_load_async_to_lds_b128 %0, %1, off" :: "v"(l0 +  32u), "v"(gA + 16384u) : "memory");
    asm volatile("global_load_async_to_lds_b128 %0, %1, off" :: "v"(l0 +  48u), "v"(gA + 16512u) : "memory");
    asm volatile("global_load_async_to_lds_b128 %0, %1, off" :: "v"(l0 +  64u), "v"(gB)          : "memory");
    asm volatile("global_load_async_to_lds_b128 %0, %1, off" :: "v"(l0 +  80u), "v"(gB + 128u)   : "memory");
    asm volatile("global_load_async_to_lds_b128 %0, %1, off" :: "v"(l0 +  96u), "v"(gB + 16384u) : "memory");
    asm volatile("global_load_async_to_lds_b128 %0, %1, off" :: "v"(l0 + 112u), "v"(gB + 16512u) : "memory");
    asm volatile("s_wait_asynccnt 0" ::: "memory");
    // Each thread reads only its own slots: no workgroup barrier needed.

    // output strides (elements): W=1, H=64, D=4096, CH=131072, B=16777216
    const unsigned ob = ((b * 128u + c) * 32u + d) * 4096u + h * 64u + 2u * wq;

    haar_butterfly_store(slot[0], slot[1], slot[2], slot[3], out + ob);           // h
    haar_butterfly_store(slot[4], slot[5], slot[6], slot[7], out + ob + 2048u);   // h+32
}

extern "C" void kernel_launch(void* const* d_in, const int* in_sizes, int n_in,
                              void* d_out, int out_size, void* d_ws, size_t ws_size,
                              hipStream_t stream) {
    const float* x = (const float*)d_in[0];
    float* out = (float*)d_out;

    // out_size = 8*128*32*64*64 = 134,217,728; each thread produces 32 outputs
    const int total_threads = out_size / 32;            // 4,194,304
    const int blocks = total_threads / THREADS;         // 16,384 (exact)

    haar3_fwd_kernel<<<blocks, THREADS, 0, stream>>>(x, out);
}